// DynamicNeuralTuringMachine_37022618091850
// MI455X (gfx1250) — compile-verified
//
#include <hip/hip_runtime.h>
#include <math.h>

// ---------------- problem dims (fixed by the reference) ----------------
#define CDIM   128   // memory content width
#define ADIM   24    // address width
#define HDIM   256   // hidden
#define XDIM   128   // input
#define QDIM   152   // A + C
#define G3H    768   // 3*H
#define NSTEPS 4

// ---------------- kernel tiling ----------------
#define TILE   64            // rows per LDS tile
#define BLOCK  128           // threads per pass block (4 waves of 32)
#define ROWPAD 132           // 128 + 4  -> conflict-free column access
#define MAPAD  25            // 24 + 1   -> conflict-free
#define PSTRIDE 130          // per-block partial: [max, sum, content[128]]
#define MAXNB  2048

// ---------------- state layout in workspace (floats) ----------------
#define IDX_H        0       // 256
#define IDX_QA       256     // 24
#define IDX_QC       280     // 128
#define IDX_BETA     408
#define IDX_M        409
#define IDX_INVZ     410
#define IDX_BETAPREV 411
#define IDX_ERASE    416     // 128
#define IDX_CAND     544     // 128
#define STATE_F      704

#if defined(__gfx1250__) && __has_builtin(__builtin_amdgcn_wmma_f32_16x16x4_f32)
#define USE_WMMA 1
typedef __attribute__((ext_vector_type(2))) float v2f;
typedef __attribute__((ext_vector_type(8))) float v8f;
#endif

// Async global->LDS staging (CDNA5 ASYNCcnt path); fall back to the proven
// b128-load + ds_store path if the builtins are not exposed.
// Probe result (round 2 diagnostic): param 1 is
//   "__attribute__((__vector_size__(4*sizeof(int)))) int __device__ *"
// i.e. an address_space(1) pointer to a vector_size(16) int — cast accordingly.
#if defined(__gfx1250__) && __has_builtin(__builtin_amdgcn_global_load_async_to_lds_b128) && __has_builtin(__builtin_amdgcn_s_wait_asynccnt)
#define USE_ASYNC_LDS 1
typedef int v4i_vs __attribute__((vector_size(16)));
#define AS1CAST(p) ((__attribute__((address_space(1))) v4i_vs*)(p))
#define AS3CAST(p) ((__attribute__((address_space(3))) v4i_vs*)(p))
#endif

__device__ __forceinline__ float sigm(float v) { return 1.f / (1.f + __expf(-v)); }

// =====================================================================
// k_init: q1, beta1 from h0; store h0 into state.
// =====================================================================
__global__ __launch_bounds__(256)
void k_init(const float* __restrict__ h0,
            const float* __restrict__ Wq, const float* __restrict__ bq,
            const float* __restrict__ u,  const float* __restrict__ bs,
            float* __restrict__ state) {
  __shared__ float sh[HDIM];
  const int t = threadIdx.x;
  sh[t] = h0[t];
  state[IDX_H + t] = sh[t];
  __syncthreads();
  if (t < QDIM) {
    float q = bq[t];
    for (int k = 0; k < HDIM; ++k) q += sh[k] * Wq[(size_t)k * QDIM + t];
    if (t < ADIM) state[IDX_QA + t] = q;
    else          state[IDX_QC + (t - ADIM)] = q;
  }
  if (t == 0) {
    float d = bs[0];
    for (int k = 0; k < HDIM; ++k) d += sh[k] * u[k];
    state[IDX_BETA] = logf(1.f + __expf(d)) + 1.f;   // softplus + 1
  }
}

// =====================================================================
// k_pass: fused [apply previous write] + [sim via WMMA] + [flash content]
//   UPDATE: apply mem update of the previous step (w from stored sims)
//   WRITE : write updated mem back to global (skipped on the last step)
// =====================================================================
template <bool UPDATE, bool WRITE>
__global__ __launch_bounds__(BLOCK)
void k_pass(const float* __restrict__ memIn, float* __restrict__ memOut,
            const float* __restrict__ MA,
            float* __restrict__ sBuf,           // raw sims, read(prev)/write(new)
            const float* __restrict__ state,
            float* __restrict__ partials,
            int nRows) {
  __shared__ float sMem[TILE * ROWPAD];   // 33.8 KB
  __shared__ float sMA [TILE * MAPAD];    //  6.4 KB
  __shared__ float sQc [CDIM];
  __shared__ float sQa [ADIM];
  __shared__ float sSim[TILE];
  __shared__ float sEw [TILE];

  const int t   = threadIdx.x;
  const int bid = blockIdx.x;
  const float beta = state[IDX_BETA];

  float betaPrev = 0.f, Mprev = 0.f, invZ = 0.f, eCol = 0.f, cdCol = 0.f;
  if (UPDATE) {
    betaPrev = state[IDX_BETAPREV];
    Mprev    = state[IDX_M];
    invZ     = state[IDX_INVZ];
    eCol     = state[IDX_ERASE + t];
    cdCol    = state[IDX_CAND + t];
  }
  sQc[t] = state[IDX_QC + t];
  if (t < ADIM) sQa[t] = state[IDX_QA + t];

  float runM = -__builtin_inff();
  float runZ = 0.f, runAcc = 0.f;
  const int nTiles = (nRows + TILE - 1) / TILE;

  for (int tile = bid; tile < nTiles; tile += gridDim.x) {
    const int row0 = tile * TILE;
    const int rows = min(TILE, nRows - row0);

    // prefetch this block's next tile into the cache hierarchy
    {
      long long nr0 = (long long)(tile + gridDim.x) * TILE;
      if (nr0 < nRows) {
        __builtin_prefetch(memIn + nr0 * CDIM + t * 64, 0, 1);     // 128*256B = whole tile
        if (t < 24) __builtin_prefetch(MA + nr0 * ADIM + t * 64, 0, 1);
      }
    }

    // ---- mem tile -> LDS ----
#ifdef USE_ASYNC_LDS
    for (int i = t; i < TILE * 32; i += BLOCK) {
      const int r = i >> 5, c4 = (i & 31) << 2;
      if (r < rows) {
        __builtin_amdgcn_global_load_async_to_lds_b128(
            AS1CAST(memIn + (size_t)(row0 + r) * CDIM + c4),
            AS3CAST(&sMem[r * ROWPAD + c4]), 0, 0);
      } else {
        *(float4*)(&sMem[r * ROWPAD + c4]) = make_float4(0.f, 0.f, 0.f, 0.f);
      }
    }
#else
    for (int i = t; i < TILE * 32; i += BLOCK) {
      const int r = i >> 5, c4 = (i & 31) << 2;
      float4 v;
      if (r < rows) v = *(const float4*)(memIn + (size_t)(row0 + r) * CDIM + c4);
      else          v = make_float4(0.f, 0.f, 0.f, 0.f);
      *(float4*)(&sMem[r * ROWPAD + c4]) = v;
    }
#endif
    // ---- addresses (contiguous copy, re-padded in LDS) ----
    for (int i = t; i < TILE * ADIM; i += BLOCK) {
      const int r = i / ADIM, c = i - r * ADIM;
      sMA[r * MAPAD + c] = (r < rows) ? MA[(size_t)row0 * ADIM + i] : 0.f;
    }
    // ---- previous-step write weights ----
    if (UPDATE && t < TILE) {
      float w = 0.f;
      if (t < rows) w = __expf(betaPrev * sBuf[row0 + t] - Mprev) * invZ;
      sEw[t] = w;
    }
#ifdef USE_ASYNC_LDS
    __builtin_amdgcn_s_wait_asynccnt(0);
#endif
    __syncthreads();

    // ---- apply previous step's erase/candidate write (in LDS + global) ----
    if (UPDATE) {
      for (int r = 0; r < rows; ++r) {
        const float w = sEw[r];
        float m = sMem[r * ROWPAD + t];
        m = m * (1.f - w * eCol) + w * cdCol;
        sMem[r * ROWPAD + t] = m;
        if (WRITE) memOut[(size_t)(row0 + r) * CDIM + t] = m;
      }
      __syncthreads();
    }

    // ---- sims: each wave does 16 rows with V_WMMA_F32_16X16X4_F32 ----
#ifdef USE_WMMA
    {
      const int wv = t >> 5, lane = t & 31;
      const int rowA = wv * 16 + (lane & 15);     // A layout: lanes 0-15 = M, halves split K
      const int koff = (lane >> 4) * 2;           // lanes 16-31 hold K+2,K+3
      const float* ar = &sMem[rowA * ROWPAD];
      v8f acc = {};
      for (int kb = 0; kb < CDIM; kb += 4) {
        v2f a = { ar[kb + koff], ar[kb + koff + 1] };
        v2f b = { sQc[kb + koff], sQc[kb + koff + 1] };   // q broadcast to all N columns
        acc = __builtin_amdgcn_wmma_f32_16x16x4_f32(false, a, false, b,
                                                    (short)0, acc, false, false);
      }
      const float* mr = &sMA[rowA * MAPAD];
      for (int kb = 0; kb < ADIM; kb += 4) {
        v2f a = { mr[kb + koff], mr[kb + koff + 1] };
        v2f b = { sQa[kb + koff], sQa[kb + koff + 1] };
        acc = __builtin_amdgcn_wmma_f32_16x16x4_f32(false, a, false, b,
                                                    (short)0, acc, false, false);
      }
      // D: lane 0 holds M=0..7 in v0..7, lane 16 holds M=8..15 (all columns equal)
      if ((lane & 15) == 0) {
        const int rbase = wv * 16 + (lane >> 4) * 8;
        #pragma unroll
        for (int v = 0; v < 8; ++v) sSim[rbase + v] = acc[v];
      }
    }
#else
    if (t < TILE) {
      float s = 0.f;
      const float* rp = &sMem[t * ROWPAD];
      for (int c = 0; c < CDIM; ++c) s += rp[c] * sQc[c];
      const float* mp = &sMA[t * MAPAD];
      for (int a = 0; a < ADIM; ++a) s += mp[a] * sQa[a];
      sSim[t] = s;
    }
#endif
    __syncthreads();

    if (t < rows) sBuf[row0 + t] = sSim[t];     // raw sims for next pass's w

    // ---- flash-softmax partial update (deterministic, per-block) ----
    float tm = -__builtin_inff();
    for (int r = 0; r < rows; ++r) tm = fmaxf(tm, sSim[r]);
    const float newM  = fmaxf(runM, beta * tm);         // beta > 1 always
    const float scale = __expf(runM - newM);
    runZ *= scale; runAcc *= scale; runM = newM;

    if (t < TILE) {
      sEw[t] = (t < rows) ? __expf(beta * sSim[t] - newM) : 0.f;
    }
    __syncthreads();
    for (int r = 0; r < rows; ++r) {
      const float e = sEw[r];
      runZ   += e;                               // identical across threads
      runAcc += e * sMem[r * ROWPAD + t];        // column t of the content read
    }
    __syncthreads();
  }

  partials[(size_t)bid * PSTRIDE + 2 + t] = runAcc;
  if (t == 0) {
    partials[(size_t)bid * PSTRIDE]     = runM;
    partials[(size_t)bid * PSTRIDE + 1] = runZ;
  }
}

// =====================================================================
// k_combine: deterministic fixed-order reduction + GRU controller + next
//            step's query/beta/erase/cand.  Single block of 256 threads.
// =====================================================================
__global__ __launch_bounds__(256)
void k_combine(const float* __restrict__ partials, int NB,
               const float* __restrict__ x,
               const float* __restrict__ Wih, const float* __restrict__ Whh,
               const float* __restrict__ bih, const float* __restrict__ bhh,
               const float* __restrict__ We,  const float* __restrict__ be,
               const float* __restrict__ Wch, const float* __restrict__ Wcx,
               const float* __restrict__ bc,
               const float* __restrict__ Wq,  const float* __restrict__ bq,
               const float* __restrict__ u,   const float* __restrict__ bs,
               float* __restrict__ state,
               float* __restrict__ out)       // non-null only on the final step
{
  __shared__ float sh  [HDIM];
  __shared__ float sInp[HDIM];     // [x | content]
  __shared__ float sGi [G3H];
  __shared__ float sGh [G3H];
  __shared__ float sHn [HDIM];
  __shared__ float sRed[256];
  __shared__ float sScale[MAXNB];

  const int t = threadIdx.x;
  const float betaCur = state[IDX_BETA];
  sh[t] = state[IDX_H + t];

  // global max of beta*sim
  float lm = -__builtin_inff();
  for (int b = t; b < NB; b += 256) lm = fmaxf(lm, partials[(size_t)b * PSTRIDE]);
  sRed[t] = lm; __syncthreads();
  for (int s = 128; s > 0; s >>= 1) { if (t < s) sRed[t] = fmaxf(sRed[t], sRed[t + s]); __syncthreads(); }
  const float M = sRed[0];
  __syncthreads();

  for (int b = t; b < NB; b += 256) sScale[b] = __expf(partials[(size_t)b * PSTRIDE] - M);
  __syncthreads();

  // partition function Z
  float lz = 0.f;
  for (int b = t; b < NB; b += 256) lz += partials[(size_t)b * PSTRIDE + 1] * sScale[b];
  sRed[t] = lz; __syncthreads();
  for (int s = 128; s > 0; s >>= 1) { if (t < s) sRed[t] += sRed[t + s]; __syncthreads(); }
  const float invZ = 1.f / sRed[0];
  __syncthreads();

  // inp = [x, content]
  if (t < XDIM) {
    sInp[t] = x[t];
  } else {
    const int c = t - XDIM;
    float acc = 0.f;
    for (int b = 0; b < NB; ++b) acc += partials[(size_t)b * PSTRIDE + 2 + c] * sScale[b];
    sInp[t] = acc * invZ;
  }
  __syncthreads();

  // GRU gates (X+C == H == 256)
  for (int jj = 0; jj < 3; ++jj) {
    const int j = t + jj * 256;
    float gi = bih[j], gh = bhh[j];
    for (int k = 0; k < HDIM; ++k) {
      gi += sInp[k] * Wih[(size_t)k * G3H + j];
      gh += sh[k]   * Whh[(size_t)k * G3H + j];
    }
    sGi[j] = gi; sGh[j] = gh;
  }
  __syncthreads();

  const float r  = sigm(sGi[t] + sGh[t]);
  const float zg = sigm(sGi[256 + t] + sGh[256 + t]);
  const float n  = tanhf(sGi[512 + t] + r * sGh[512 + t]);
  const float hn = (1.f - zg) * n + zg * sh[t];
  sHn[t] = hn;
  state[IDX_H + t] = hn;

  if (out != nullptr) { out[t] = hn; return; }   // uniform branch: final step
  __syncthreads();

  if (t < CDIM) {
    float ev = be[t], cv = bc[t];
    for (int k = 0; k < HDIM; ++k) {
      ev += sHn[k] * We [(size_t)k * CDIM + t];
      cv += sHn[k] * Wch[(size_t)k * CDIM + t];
    }
    for (int k = 0; k < XDIM; ++k) cv += sInp[k] * Wcx[(size_t)k * CDIM + t];
    state[IDX_ERASE + t] = sigm(ev);
    state[IDX_CAND  + t] = fmaxf(cv, 0.f);
  }
  if (t < QDIM) {
    float q = bq[t];
    for (int k = 0; k < HDIM; ++k) q += sHn[k] * Wq[(size_t)k * QDIM + t];
    if (t < ADIM) state[IDX_QA + t] = q;
    else          state[IDX_QC + (t - ADIM)] = q;
  }
  if (t == 255) {
    float d = bs[0];
    for (int k = 0; k < HDIM; ++k) d += sHn[k] * u[k];
    state[IDX_BETA]     = logf(1.f + __expf(d)) + 1.f;
    state[IDX_M]        = M;
    state[IDX_INVZ]     = invZ;
    state[IDX_BETAPREV] = betaCur;
  }
}

// =====================================================================
// Host launcher.  Workspace layout (floats):
//   memW[N*128] | sims[N] | partials[NB*130] | state[1024]
//   (~258 MB for N = 500k; ws is assumed sized for the reference shapes)
// =====================================================================
extern "C" void kernel_launch(void* const* d_in, const int* in_sizes, int n_in,
                              void* d_out, int out_size, void* d_ws, size_t ws_size,
                              hipStream_t stream) {
  const float* x   = (const float*)d_in[0];
  const float* h0  = (const float*)d_in[1];
  const float* mem = (const float*)d_in[2];
  const float* ma  = (const float*)d_in[3];
  const float* Wq  = (const float*)d_in[4];
  const float* bq  = (const float*)d_in[5];
  const float* u   = (const float*)d_in[6];
  const float* bs  = (const float*)d_in[7];
  const float* We  = (const float*)d_in[8];
  const float* be  = (const float*)d_in[9];
  const float* Wch = (const float*)d_in[10];
  const float* Wcx = (const float*)d_in[11];
  const float* bc  = (const float*)d_in[12];
  const float* Wih = (const float*)d_in[13];
  const float* Whh = (const float*)d_in[14];
  const float* bih = (const float*)d_in[15];
  const float* bhh = (const float*)d_in[16];
  // d_in[17] = num_addressing_steps (device scalar); reference fixes it at 4.

  const int N = in_sizes[2] / CDIM;
  int NB = (N + TILE - 1) / TILE;
  if (NB > MAXNB) NB = MAXNB;

  float* ws    = (float*)d_ws;
  float* memW  = ws;
  float* sims  = memW + (size_t)N * CDIM;
  float* part  = sims + N;
  float* state = part + (size_t)NB * PSTRIDE;
  float* out   = (float*)d_out;

  k_init<<<1, 256, 0, stream>>>(h0, Wq, bq, u, bs, state);

  // step 1: sims/content over original memory (no update yet)
  k_pass<false, false><<<NB, BLOCK, 0, stream>>>(mem, memW, ma, sims, state, part, N);
  k_combine<<<1, 256, 0, stream>>>(part, NB, x, Wih, Whh, bih, bhh,
                                   We, be, Wch, Wcx, bc, Wq, bq, u, bs, state, nullptr);
  // step 2: apply write #1 (orig -> memW), sims/content for step 2
  k_pass<true, true><<<NB, BLOCK, 0, stream>>>(mem, memW, ma, sims, state, part, N);
  k_combine<<<1, 256, 0, stream>>>(part, NB, x, Wih, Whh, bih, bhh,
                                   We, be, Wch, Wcx, bc, Wq, bq, u, bs, state, nullptr);
  // step 3: apply write #2 (memW -> memW), sims/content for step 3
  k_pass<true, true><<<NB, BLOCK, 0, stream>>>(memW, memW, ma, sims, state, part, N);
  k_combine<<<1, 256, 0, stream>>>(part, NB, x, Wih, Whh, bih, bhh,
                                   We, be, Wch, Wcx, bc, Wq, bq, u, bs, state, nullptr);
  // step 4: apply write #3 in-register only (no global write-back needed)
  k_pass<true, false><<<NB, BLOCK, 0, stream>>>(memW, memW, ma, sims, state, part, N);
  k_combine<<<1, 256, 0, stream>>>(part, NB, x, Wih, Whh, bih, bhh,
                                   We, be, Wch, Wcx, bc, Wq, bq, u, bs, state, out);
}